// Conv1DCppnVar_54520314855646
// MI455X (gfx1250) — compile-verified
//
#include <hip/hip_runtime.h>

// Conv1D with per-output-position weights ("CPPN" conv).
// Memory-bound: w (536 MB fp32) streamed once from HBM -> ~23 us floor @ 23.3 TB/s.
// x (16.8 MB) is fully L2-resident, so per-o window re-reads are free.
// Compute mapped to V_WMMA_F32_16X16X4_F32 (exact fp32): per o, a 16x32x512 GEMM
// (M=b, N=d, K=c*16+k), one wave per o, 2 accumulators (two 16-wide d tiles),
// 128 chained K-steps. 4 consecutive o per workgroup => adjacent waves share
// the two 64B halves of each 128B w cacheline.

typedef float v2f __attribute__((ext_vector_type(2)));
typedef float v8f __attribute__((ext_vector_type(8)));

#define BB    16
#define CIN   32
#define COUT  32
#define KW    16
#define INS   8192
#define OUTS  8176

__global__ __launch_bounds__(128) void conv1d_cppn_wmma(
    const float* __restrict__ x,
    const float* __restrict__ w,
    const float* __restrict__ bias,
    float* __restrict__ out)
{
    const int lane = threadIdx.x & 31;
    const int wave = threadIdx.x >> 5;
    const int l16  = lane & 15;
    const int hi   = lane >> 4;                // 0: K=0,1 ; 1: K=2,3 (A and B layouts)
    const int o    = blockIdx.x * 4 + wave;    // one output position per wave

    v8f acc0 = {};   // d = 0..15
    v8f acc1 = {};   // d = 16..31

    // A fragment source: x[b=l16][c][o + k]
    const float* xrow = x + (size_t)l16 * CIN * INS + o;
    // B fragment source: w[d][c][o][k], d = l16 (tile0) / l16+16 (tile1)
    const size_t wcs = (size_t)OUTS * KW;                       // c stride (elements)
    const float* wp0 = w + ((size_t)(l16)      * CIN * OUTS + o) * KW;
    const float* wp1 = w + ((size_t)(l16 + 16) * CIN * OUTS + o) * KW;

    for (int c = 0; c < CIN; ++c) {
        const float* xp = xrow + (size_t)c * INS;
        const float* q0 = wp0 + (size_t)c * wcs;
        const float* q1 = wp1 + (size_t)c * wcs;
#pragma unroll
        for (int j = 0; j < 4; ++j) {
            const int k0 = 4 * j + 2 * hi;
            // A 16x4 f32: lane l<16 -> (K=0,K=1), lane l+16 -> (K=2,K=3)
            v2f a;
            a.x = xp[k0];
            a.y = xp[k0 + 1];
            // B 4x16 f32: same K split across lane halves; 8B-aligned (o-stride=64B)
            v2f b0 = *(const v2f*)(q0 + k0);
            v2f b1 = *(const v2f*)(q1 + k0);
            acc0 = __builtin_amdgcn_wmma_f32_16x16x4_f32(
                false, a, false, b0, (short)0, acc0, false, false);
            acc1 = __builtin_amdgcn_wmma_f32_16x16x4_f32(
                false, a, false, b1, (short)0, acc1, false, false);
        }
    }

    // D layout: lane l16 = N = d (within tile), VGPR v + 8*hi = M = b
    const float bv0 = bias[l16];
    const float bv1 = bias[16 + l16];
#pragma unroll
    for (int v = 0; v < 8; ++v) {
        const int b = v + 8 * hi;
        float r0 = acc0[v] + bv0; r0 = r0 > 0.f ? r0 : 0.f;
        float r1 = acc1[v] + bv1; r1 = r1 > 0.f ? r1 : 0.f;
        out[((size_t)b * COUT + l16)        * OUTS + o] = r0;
        out[((size_t)b * COUT + 16 + l16)   * OUTS + o] = r1;
    }
}

extern "C" void kernel_launch(void* const* d_in, const int* in_sizes, int n_in,
                              void* d_out, int out_size, void* d_ws, size_t ws_size,
                              hipStream_t stream) {
    (void)in_sizes; (void)n_in; (void)out_size; (void)d_ws; (void)ws_size;
    const float* x    = (const float*)d_in[0];
    const float* w    = (const float*)d_in[1];
    const float* bias = (const float*)d_in[2];
    float* out        = (float*)d_out;

    dim3 grid(OUTS / 4);   // 2044 workgroups, 4 waves each, one o per wave
    dim3 block(128);
    hipLaunchKernelGGL(conv1d_cppn_wmma, grid, block, 0, stream, x, w, bias, out);
}